// Set2Graph_57234734186752
// MI455X (gfx1250) — compile-verified
//
#include <hip/hip_runtime.h>

#define D_FEAT 16
#define D_HID  64
#define D_COM  8
#define D_EMB  8
#define NB     512
#define NN     25
#define NT     (NB * NN)        // 12800 nodes
#define EPG    (NN * (NN - 1))  // 600 edges per graph
#define NODE_OUT (NT * 4)       // 51200 floats, edge preds follow

typedef __attribute__((ext_vector_type(2))) float v2f;
typedef __attribute__((ext_vector_type(8))) float v8f;

// D = A(16x4 f32) * B(4x16 f32) + C ; full-fp32 CDNA5 WMMA
__device__ __forceinline__ v8f wmma_f32(v8f c, v2f a, v2f b) {
    return __builtin_amdgcn_wmma_f32_16x16x4_f32(
        /*neg_a=*/false, a, /*neg_b=*/false, b,
        /*c_mod=*/(short)0, c, /*reuse_a=*/false, /*reuse_b=*/false);
}

__device__ __forceinline__ v2f lds_v2(const float* p) {
    return *(const v2f*)p;   // contiguous pair -> ds_load_b64
}

// ---------------------------------------------------------------------------
// Kernel 1: per-graph mean of node_hidden -> workspace [NB][64]
// ---------------------------------------------------------------------------
__global__ void __launch_bounds__(64) mean_kernel(const float* __restrict__ hid,
                                                  float* __restrict__ meanh) {
    const int g = blockIdx.x;
    const int j = threadIdx.x;  // 0..63
    const float* p = hid + (size_t)g * NN * D_HID + j;
    float s = 0.f;
#pragma unroll
    for (int i = 0; i < NN; ++i) s += p[i * D_HID];
    meanh[g * D_HID + j] = s * (1.0f / (float)NN);
}

// ---------------------------------------------------------------------------
// Kernel 2: node MLP, 64 nodes/block: 160 -> 128 -> relu(64) -> 4
// Weights staged transposed in LDS; all WMMA fragments come from LDS.
// ---------------------------------------------------------------------------
__global__ void __launch_bounds__(256) node_mlp_kernel(
    const float* __restrict__ feat, const float* __restrict__ hid,
    const float* __restrict__ com,  const float* __restrict__ emb,
    const float* __restrict__ w0, const float* __restrict__ b0,
    const float* __restrict__ w1, const float* __restrict__ b1,
    const float* __restrict__ w2, const float* __restrict__ b2,
    const float* __restrict__ meanh, float* __restrict__ out)
{
    __shared__ float W0t[128][164];  // [n][k], pad->stride%64==36 (conflict-free)
    __shared__ float W1t[64][132];   // [n][k]
    __shared__ float Als[64][164];   // node_in rows
    __shared__ float H0 [64][132];
    __shared__ float H1 [64][68];
    __shared__ float W2s[256];       // 64x4

    const int t    = threadIdx.x;
    const int wave = t >> 5;
    const int lane = t & 31;
    const int half = lane >> 4;
    const int l16  = lane & 15;
    const int node0 = blockIdx.x * 64;

    // stage transposed weights (coalesced global reads)
    for (int p = t; p < 160 * 128; p += 256) W0t[p & 127][p >> 7] = w0[p];
    for (int p = t; p < 128 * 64;  p += 256) W1t[p & 63][p >> 6]  = w1[p];
    W2s[t] = w2[t];

    // gather node_in = [feat | hid | mean_hidden | emb | com]
    for (int p = t; p < 64 * 160; p += 256) {
        const int m = p / 160, c = p % 160;
        const int node = node0 + m;
        float v;
        if      (c < 16)  v = feat[node * 16 + c];
        else if (c < 80)  v = hid[node * 64 + (c - 16)];
        else if (c < 144) v = meanh[(node / NN) * 64 + (c - 80)];
        else if (c < 152) v = emb[node * 8 + (c - 144)];
        else              v = com[node * 8 + (c - 152)];
        Als[m][c] = v;
    }
    __syncthreads();

    // layer 0: [64,160] @ [160,128] + b0 ; 32 tile-tasks, 4 per wave
    for (int tt = wave; tt < 32; tt += 8) {
        const int mt = tt >> 3, nb = (tt & 7) * 16;
        v8f c = {};
#pragma unroll 4
        for (int k0 = 0; k0 < 160; k0 += 4) {
            const int kk = k0 + 2 * half;
            v2f a = lds_v2(&Als[mt * 16 + l16][kk]);
            v2f b = lds_v2(&W0t[nb + l16][kk]);
            c = wmma_f32(c, a, b);
        }
        const float bias = b0[nb + l16];
#pragma unroll
        for (int i = 0; i < 8; ++i)
            H0[mt * 16 + i + 8 * half][nb + l16] = c[i] + bias;
    }
    __syncthreads();

    // layer 1: relu([64,128] @ [128,64] + b1) ; 16 tile-tasks, 2 per wave
    for (int tt = wave; tt < 16; tt += 8) {
        const int mt = tt >> 2, nb = (tt & 3) * 16;
        v8f c = {};
#pragma unroll 4
        for (int k0 = 0; k0 < 128; k0 += 4) {
            const int kk = k0 + 2 * half;
            v2f a = lds_v2(&H0[mt * 16 + l16][kk]);
            v2f b = lds_v2(&W1t[nb + l16][kk]);
            c = wmma_f32(c, a, b);
        }
        const float bias = b1[nb + l16];
#pragma unroll
        for (int i = 0; i < 8; ++i)
            H1[mt * 16 + i + 8 * half][nb + l16] = fmaxf(c[i] + bias, 0.f);
    }
    __syncthreads();

    // layer 2: [64,64] @ [64,4] + b2 ; one dot per thread
    {
        const int m = t >> 2, cls = t & 3;
        float acc = b2[cls];
#pragma unroll 8
        for (int k = 0; k < 64; ++k) acc += H1[m][k] * W2s[k * 4 + cls];
        out[(node0 + m) * 4 + cls] = acc;
    }
}

// ---------------------------------------------------------------------------
// Kernel 3: edge MLP, one workgroup per graph.
//   h0(s,d) = P[s] + Q[d] + R   (layer-0 factorization, 24x flop cut)
//   W1 staged transposed in LDS -> hot loop is pure LDS + WMMA.
// ---------------------------------------------------------------------------
__global__ void __launch_bounds__(256) edge_mlp_kernel(
    const float* __restrict__ feat, const float* __restrict__ hid,
    const float* __restrict__ com,  const float* __restrict__ emb,
    const float* __restrict__ w0, const float* __restrict__ b0,
    const float* __restrict__ w1, const float* __restrict__ b1,
    const float* __restrict__ w2, const float* __restrict__ b2,
    const float* __restrict__ meanh, float* __restrict__ out)
{
    __shared__ float A  [32][100];   // [feat|hid|com|emb], rows>=25 zero; padded
    __shared__ float P  [NN][256];   // src-part @ W0
    __shared__ float Q  [NN][256];   // dst-part @ W0
    __shared__ float R  [256];       // mean-part @ W0 + b0
    __shared__ float Mh [64];
    __shared__ float W1t[128][260];  // [n][k] transposed W1, padded
    __shared__ float H0 [16][260];   // padded (A-source, conflict-free)
    __shared__ float H1 [16][132];
    __shared__ float W2s[128];
    __shared__ float S  [16][17];    // layer-2 partial sums

    const int t    = threadIdx.x;
    const int wave = t >> 5;
    const int lane = t & 31;
    const int half = lane >> 4;
    const int l16  = lane & 15;
    const int g    = blockIdx.x;
    const int node_base = g * NN;

    // stage transposed W1 (coalesced reads) + w2
    for (int p = t; p < 256 * 128; p += 256) W1t[p & 127][p >> 7] = w1[p];
    if (t < 128) W2s[t] = w2[t];

    // stage node features (zero the 7 pad rows)
    for (int p = t; p < 32 * 96; p += 256) {
        const int m = p / 96, c = p % 96;
        float v = 0.f;
        if (m < NN) {
            const int node = node_base + m;
            if      (c < 16) v = feat[node * 16 + c];
            else if (c < 80) v = hid[node * 64 + (c - 16)];
            else if (c < 88) v = com[node * 8 + (c - 80)];
            else             v = emb[node * 8 + (c - 88)];
        }
        A[m][c] = v;
    }
    if (t < 64) Mh[t] = meanh[g * 64 + t];
    __syncthreads();

    // R[n] = b0[n] + sum_k mean[k] * W0[192+k][n]
    {
        float acc = b0[t];
#pragma unroll 8
        for (int k = 0; k < 64; ++k) acc += Mh[k] * w0[(192 + k) * 256 + t];
        R[t] = acc;
    }

    // P,Q via WMMA: 2 matrices x 2 M-tiles x 16 N-tiles = 64 tasks, 8/wave
    for (int tt = wave; tt < 64; tt += 8) {
        const int sel = tt >> 5;          // 0 = P (src rows), 1 = Q (dst rows)
        const int mt  = (tt >> 4) & 1;
        const int nb  = (tt & 15) * 16;
        v8f c = {};
#pragma unroll 4
        for (int k0 = 0; k0 < 96; k0 += 4) {
            const int kk = k0 + 2 * half;
            const int k1 = kk + 1;
            v2f a = lds_v2(&A[mt * 16 + l16][kk]);
            int r0, r1;
            if (sel == 0) {  // src slices at edge_in rows 0..87, 184..191
                r0 = (kk < 88) ? kk : kk + 96;
                r1 = (k1 < 88) ? k1 : k1 + 96;
            } else {         // dst slices at edge_in rows 96.., 88.., 176..
                r0 = (kk < 80) ? kk + 96 : ((kk < 88) ? kk + 8 : kk + 88);
                r1 = (k1 < 80) ? k1 + 96 : ((k1 < 88) ? k1 + 8 : k1 + 88);
            }
            v2f b;
            b.x = w0[r0 * 256 + nb + l16];
            b.y = w0[r1 * 256 + nb + l16];
            c = wmma_f32(c, a, b);
        }
#pragma unroll
        for (int i = 0; i < 8; ++i) {
            const int row = mt * 16 + i + 8 * half;
            if (row < NN) {
                if (sel == 0) P[row][nb + l16] = c[i];
                else          Q[row][nb + l16] = c[i];
            }
        }
    }
    __syncthreads();

    const float bias1 = b1[wave * 16 + l16];   // hoisted out of tile loop

    // 38 tiles of 16 edges
    const int ntiles = (EPG + 15) / 16;
    for (int et = 0; et < ntiles; ++et) {
        // H0[m][n] = P[s_m][n] + Q[d_m][n] + R[n]
#pragma unroll
        for (int i = 0; i < 16; ++i) {
            const int e = et * 16 + i;
            float v = 0.f;
            if (e < EPG) {
                const int s = e / 24;
                const int r = e % 24;
                const int d = r + (r >= s ? 1 : 0);
                v = P[s][t] + Q[d][t] + R[t];
            }
            H0[i][t] = v;
        }
        __syncthreads();

        // layer 1: relu([16,256] @ [256,128] + b1) ; pure LDS hot loop
        {
            const int nb = wave * 16;
            v8f c = {};
#pragma unroll 4
            for (int k0 = 0; k0 < 256; k0 += 4) {
                const int kk = k0 + 2 * half;
                v2f a = lds_v2(&H0[l16][kk]);
                v2f b = lds_v2(&W1t[nb + l16][kk]);
                c = wmma_f32(c, a, b);
            }
#pragma unroll
            for (int i = 0; i < 8; ++i)
                H1[i + 8 * half][nb + l16] = fmaxf(c[i] + bias1, 0.f);
        }
        __syncthreads();

        // layer 2 partials: 16 threads per edge, 8 MACs each
        {
            const int m = t >> 4, j = t & 15;
            float acc = 0.f;
#pragma unroll
            for (int q = 0; q < 8; ++q)
                acc += H1[m][j * 8 + q] * W2s[j * 8 + q];
            S[m][j] = acc;
        }
        __syncthreads();

        if (t < 16) {
            const int e = et * 16 + t;
            if (e < EPG) {
                float acc = b2[0];
#pragma unroll
                for (int j = 0; j < 16; ++j) acc += S[t][j];
                out[NODE_OUT + g * EPG + e] = acc;
            }
        }
        __syncthreads();
    }
}

// ---------------------------------------------------------------------------
extern "C" void kernel_launch(void* const* d_in, const int* in_sizes, int n_in,
                              void* d_out, int out_size, void* d_ws, size_t ws_size,
                              hipStream_t stream) {
    const float* feat = (const float*)d_in[0];
    const float* hid  = (const float*)d_in[1];
    const float* com  = (const float*)d_in[2];
    const float* emb  = (const float*)d_in[3];
    const float* ew0  = (const float*)d_in[4];
    const float* eb0  = (const float*)d_in[5];
    const float* ew1  = (const float*)d_in[6];
    const float* eb1  = (const float*)d_in[7];
    const float* ew2  = (const float*)d_in[8];
    const float* eb2  = (const float*)d_in[9];
    const float* nw0  = (const float*)d_in[10];
    const float* nb0  = (const float*)d_in[11];
    const float* nw1  = (const float*)d_in[12];
    const float* nb1  = (const float*)d_in[13];
    const float* nw2  = (const float*)d_in[14];
    const float* nb2  = (const float*)d_in[15];
    // d_in[16..18] = src/dst/graph_ids: derived arithmetically, unused.

    float* out   = (float*)d_out;
    float* meanh = (float*)d_ws;  // NB*64 floats = 128 KB

    mean_kernel<<<NB, 64, 0, stream>>>(hid, meanh);
    node_mlp_kernel<<<NT / 64, 256, 0, stream>>>(feat, hid, com, emb,
                                                 nw0, nb0, nw1, nb1, nw2, nb2,
                                                 meanh, out);
    edge_mlp_kernel<<<NB, 256, 0, stream>>>(feat, hid, com, emb,
                                            ew0, eb0, ew1, eb1, ew2, eb2,
                                            meanh, out);
}